// mamba_encoder_43774306681538
// MI455X (gfx1250) — compile-verified
//
#include <hip/hip_runtime.h>
#include <hip/hip_bf16.h>
#include <math.h>

typedef float v2f __attribute__((ext_vector_type(2)));
typedef float v8f __attribute__((ext_vector_type(8)));
typedef int   vs4i __attribute__((vector_size(16)));   // matches builtin param type

#define DEV static __device__ __forceinline__
#define AS1 __attribute__((address_space(1)))
#define AS3 __attribute__((address_space(3)))

static constexpr int BATCH  = 16;
static constexpr int SEQ    = 8192;
static constexpr int DM     = 16;   // d_model
static constexpr int DI     = 32;   // d_inner
static constexpr int DSTATE = 16;
static constexpr int NPROJ  = 33;   // dt_rank + 2*d_state
static constexpr int HM     = 32;   // mlp hidden
static constexpr int SEGT   = 64;   // timesteps per scan segment
static constexpr int SEGS   = SEQ / SEGT;  // 128 segments per batch row
static constexpr float EPS  = 1e-6f;

// ---- CDNA5 async LDS staging (ASYNCcnt path), with plain fallback ----------
#if defined(__has_builtin)
#if __has_builtin(__builtin_amdgcn_global_load_async_to_lds_b128) && \
    __has_builtin(__builtin_amdgcn_s_wait_asynccnt)
#define HAVE_ASYNC_LOAD 1
#endif
#if __has_builtin(__builtin_amdgcn_global_store_async_from_lds_b128) && \
    __has_builtin(__builtin_amdgcn_s_wait_asynccnt)
#define HAVE_ASYNC_STORE 1
#endif
#endif

// copy 16 bytes global -> LDS for this lane
DEV void ld_async16(float* lds_dst, const float* gsrc) {
#if defined(HAVE_ASYNC_LOAD)
  __builtin_amdgcn_global_load_async_to_lds_b128(
      (AS1 vs4i*)const_cast<float*>(gsrc), (AS3 vs4i*)lds_dst, 0, 0);
#else
  *(float4*)lds_dst = *(const float4*)gsrc;
#endif
}
// copy 16 bytes LDS -> global for this lane
DEV void st_async16(float* gdst, const float* lds_src) {
#if defined(HAVE_ASYNC_STORE)
  __builtin_amdgcn_global_store_async_from_lds_b128(
      (AS1 vs4i*)gdst, (AS3 vs4i*)const_cast<float*>(lds_src), 0, 0);
#else
  *(float4*)gdst = *(const float4*)lds_src;
#endif
}
DEV void wait_async_loads() {
#if defined(HAVE_ASYNC_LOAD)
  __builtin_amdgcn_s_wait_asynccnt(0);
#endif
}

// D = A(16x4,f32) * B(4x16,f32) + C(16x16,f32), fp32 matrix core path
DEV v8f wmma4(v2f a, v2f b, v8f c) {
  return __builtin_amdgcn_wmma_f32_16x16x4_f32(false, a, false, b, (short)0, c,
                                               false, false);
}

DEV float silu_f(float v)     { return v / (1.0f + __expf(-v)); }
DEV float softplus_f(float v) { return v > 20.0f ? v : log1pf(__expf(v)); }

// ---------------------------------------------------------------------------
// Kernel 1: per 16-token tile (one wave):
//   rmsnorm -> in_proj GEMM (WMMA) -> causal conv+SiLU -> x_proj GEMM (WMMA)
//   -> dt softplus; stores z, xc, dt, B, C to workspace.
// ---------------------------------------------------------------------------
__global__ void __launch_bounds__(32)
k1_front(const float* __restrict__ x, const float* __restrict__ rms1_w,
         const float* __restrict__ in_proj_w,   // [16][64]
         const float* __restrict__ conv_w,      // [32][4]
         const float* __restrict__ conv_b,      // [32]
         const float* __restrict__ x_proj_w,    // [32][33]
         const float* __restrict__ dt_proj_w,   // [1][32]
         const float* __restrict__ dt_proj_b,   // [32]
         float* __restrict__ ws_z, float* __restrict__ ws_xc,
         float* __restrict__ ws_dt, float* __restrict__ ws_B,
         float* __restrict__ ws_C)
{
  __shared__ __align__(16) float lraw[16 * 16];  // raw tile / lookback rows
  __shared__ float lx0[16][16];                  // rmsnorm'd tile
  __shared__ float lprev[3][DI];                 // conv history (3 tokens)
  __shared__ float lxz[16][64];                  // in_proj output (xm | z)
  __shared__ float lxc[16][DI];                  // conv+silu output
  __shared__ float lproj[16][NPROJ + 3];         // x_proj output (padded)

  const int lane = threadIdx.x;
  const int nn   = lane & 15;
  const int hh   = lane >> 4;
  const size_t t0   = (size_t)blockIdx.x * 16;
  const int    tloc = (int)(t0 & (SEQ - 1));

  // ---- load x tile (async to LDS) ----
  const float* xt = x + t0 * DM;
  ld_async16(&lraw[lane * 4],       xt + lane * 4);
  ld_async16(&lraw[128 + lane * 4], xt + 128 + lane * 4);
  wait_async_loads();
  __syncthreads();

  // ---- rmsnorm rows ----
  if (lane < 16) {
    float ss = 0.f;
#pragma unroll
    for (int c = 0; c < 16; ++c) { float v = lraw[lane * 16 + c]; ss += v * v; }
    float rstd = rsqrtf(ss * (1.0f / 16.0f) + EPS);
#pragma unroll
    for (int c = 0; c < 16; ++c)
      lx0[lane][c] = lraw[lane * 16 + c] * rstd * rms1_w[c];
  }
  __syncthreads();

  // ---- 3-token lookback rows (normalized, weight-scaled) into lraw ----
  if (lane < 3) {
    if (tloc == 0) {
#pragma unroll
      for (int c = 0; c < 16; ++c) lraw[lane * 16 + c] = 0.f;
    } else {
      const float* xp = x + (t0 - 3 + (size_t)lane) * DM;
      float tmp[16]; float ss = 0.f;
#pragma unroll
      for (int c = 0; c < 16; ++c) { tmp[c] = xp[c]; ss += tmp[c] * tmp[c]; }
      float rstd = rsqrtf(ss * (1.0f / 16.0f) + EPS);
#pragma unroll
      for (int c = 0; c < 16; ++c) lraw[lane * 16 + c] = tmp[c] * rstd * rms1_w[c];
    }
  }
  __syncthreads();

  // xm for the 3 lookback tokens: lane = channel d (tiny VALU)
  {
#pragma unroll
    for (int r = 0; r < 3; ++r) {
      float acc = 0.f;
#pragma unroll
      for (int k = 0; k < 16; ++k) acc += lraw[r * 16 + k] * in_proj_w[k * 64 + lane];
      lprev[r][lane] = acc;
    }
  }

  // ---- GEMM1: xz[16x64] = x0[16x16] @ in_proj_w[16x64] (16 WMMAs) ----
  v2f afr[4];
#pragma unroll
  for (int kt = 0; kt < 4; ++kt) {
    int kk = kt * 4 + hh * 2;
    afr[kt].x = lx0[nn][kk];
    afr[kt].y = lx0[nn][kk + 1];
  }
#pragma unroll
  for (int nt = 0; nt < 4; ++nt) {
    v8f c = {0.f, 0.f, 0.f, 0.f, 0.f, 0.f, 0.f, 0.f};
#pragma unroll
    for (int kt = 0; kt < 4; ++kt) {
      int kr = kt * 4 + hh * 2;
      v2f bf;
      bf.x = in_proj_w[(kr + 0) * 64 + nt * 16 + nn];
      bf.y = in_proj_w[(kr + 1) * 64 + nt * 16 + nn];
      c = wmma4(afr[kt], bf, c);
    }
#pragma unroll
    for (int r = 0; r < 8; ++r) lxz[r + 8 * hh][nt * 16 + nn] = c[r];
  }
  __syncthreads();

  // ---- store z half ----
  for (int i = lane; i < 16 * DI; i += 32) {
    int m = i >> 5, d = i & 31;
    ws_z[(t0 + m) * DI + d] = lxz[m][32 + d];
  }

  // ---- causal depthwise conv + SiLU (lane = channel d) ----
  {
    const int d = lane;
    const float cb = conv_b[d];
#pragma unroll
    for (int m = 0; m < 16; ++m) {
      float acc = cb;
#pragma unroll
      for (int k = 0; k < 4; ++k) {
        int j = m - 3 + k;
        float xv = (j >= 0) ? lxz[j][d] : lprev[j + 3][d];
        acc += conv_w[d * 4 + k] * xv;
      }
      float xcv = silu_f(acc);
      lxc[m][d] = xcv;
      ws_xc[(t0 + m) * DI + d] = xcv;
    }
  }
  __syncthreads();

  // ---- GEMM2: proj[16x33] = xc[16x32] @ x_proj_w[32x33] (24 WMMAs) ----
  v2f a2[8];
#pragma unroll
  for (int kt = 0; kt < 8; ++kt) {
    int kk = kt * 4 + hh * 2;
    a2[kt].x = lxc[nn][kk];
    a2[kt].y = lxc[nn][kk + 1];
  }
#pragma unroll
  for (int nt = 0; nt < 3; ++nt) {
    const int n = nt * 16 + nn;
    const bool ok = (n < NPROJ);
    v8f c = {0.f, 0.f, 0.f, 0.f, 0.f, 0.f, 0.f, 0.f};
#pragma unroll
    for (int kt = 0; kt < 8; ++kt) {
      int kr = kt * 4 + hh * 2;
      v2f bf;
      bf.x = ok ? x_proj_w[(kr + 0) * NPROJ + n] : 0.f;
      bf.y = ok ? x_proj_w[(kr + 1) * NPROJ + n] : 0.f;
      c = wmma4(a2[kt], bf, c);
    }
    if (ok) {
#pragma unroll
      for (int r = 0; r < 8; ++r) lproj[r + 8 * hh][n] = c[r];
    }
  }
  __syncthreads();

  // ---- dt = softplus(proj[:,0]*dt_proj_w + dt_proj_b); B,C slices ----
  {
    const int d = lane;
    const float dtw = dt_proj_w[d];
    const float dtb = dt_proj_b[d];
#pragma unroll
    for (int m = 0; m < 16; ++m)
      ws_dt[(t0 + m) * DI + d] = softplus_f(lproj[m][0] * dtw + dtb);
  }
  for (int i = lane; i < 16 * DSTATE; i += 32) {
    int m = i >> 4, s = i & 15;
    ws_B[(t0 + m) * DSTATE + s] = lproj[m][1 + s];
    ws_C[(t0 + m) * DSTATE + s] = lproj[m][17 + s];
  }
}

// ---------------------------------------------------------------------------
// Scan pass A: segment-local scan.  block = (seg, b), lane = (d, s) chain.
// h starts at 0; emits per-segment (prod dA, local h_end).
// ---------------------------------------------------------------------------
__global__ void __launch_bounds__(512)
k2a_segscan(const float* __restrict__ ws_B, const float* __restrict__ ws_dt,
            const float* __restrict__ ws_xc,
            const float* __restrict__ A_log,
            float* __restrict__ segA, float* __restrict__ segB)
{
  __shared__ __align__(16) float cB[SEGT * 16];
  __shared__ __align__(16) float cdt[SEGT * 32];
  __shared__ __align__(16) float cxc[SEGT * 32];

  const int tid = threadIdx.x;
  const int d = tid >> 4;
  const int s = tid & 15;
  const int seg = blockIdx.x;
  const int b   = blockIdx.y;

  const size_t tok0 = (size_t)b * SEQ + (size_t)seg * SEGT;
  const float* dsrc = ws_dt + tok0 * 32;
  const float* xsrc = ws_xc + tok0 * 32;
  const float* bsrc = ws_B  + tok0 * 16;

  ld_async16(&cdt[tid * 4], dsrc + tid * 4);                 // 8 KB
  ld_async16(&cxc[tid * 4], xsrc + tid * 4);                 // 8 KB
  if (tid < 256) ld_async16(&cB[tid * 4], bsrc + tid * 4);   // 4 KB
  wait_async_loads();
  __syncthreads();

  const float A = -__expf(A_log[d * 16 + s]);
  float h = 0.f, aprod = 1.f;
#pragma unroll 4
  for (int t = 0; t < SEGT; ++t) {
    float dt = cdt[t * 32 + d];
    float dA = __expf(dt * A);
    h = dA * h + (dt * cxc[t * 32 + d]) * cB[t * 16 + s];
    aprod *= dA;
  }
  const size_t idx = ((size_t)b * SEGS + seg) * 512 + tid;
  segA[idx] = aprod;
  segB[idx] = h;
}

// ---------------------------------------------------------------------------
// Scan pass B: stitch segments.  block = b, lane = (d, s) chain; serial over
// the 128 segments: hinit[seg] = true initial state of each segment.
// ---------------------------------------------------------------------------
__global__ void __launch_bounds__(512)
k2b_stitch(const float* __restrict__ segA, const float* __restrict__ segB,
           float* __restrict__ hinit)
{
  const int tid = threadIdx.x;
  const int b   = blockIdx.x;
  float H = 0.f;
  for (int seg = 0; seg < SEGS; ++seg) {
    const size_t idx = ((size_t)b * SEGS + seg) * 512 + tid;
    if (seg + 1 < SEGS)
      __builtin_prefetch(segA + idx + 512, 0, 3);
    hinit[idx] = H;
    H = segA[idx] * H + segB[idx];
  }
}

// ---------------------------------------------------------------------------
// Scan pass C: apply.  block = (seg, b); re-run segment from true h0,
// 16-lane shfl_xor reduce for y = h.C, SiLU(z) gate; overwrites xc in place.
// ---------------------------------------------------------------------------
__global__ void __launch_bounds__(512)
k2c_apply(const float* __restrict__ ws_B, const float* __restrict__ ws_C,
          const float* __restrict__ ws_dt, float* ws_xc /* in xc, out y */,
          const float* __restrict__ ws_z,
          const float* __restrict__ A_log, const float* __restrict__ D_skip,
          const float* __restrict__ hinit)
{
  __shared__ __align__(16) float cB[SEGT * 16];
  __shared__ __align__(16) float cC[SEGT * 16];
  __shared__ __align__(16) float cdt[SEGT * 32];
  __shared__ __align__(16) float cxc[SEGT * 32];
  __shared__ __align__(16) float cz[SEGT * 32];
  __shared__ __align__(16) float cy[SEGT * 32];

  const int tid = threadIdx.x;
  const int d = tid >> 4;
  const int s = tid & 15;
  const int seg = blockIdx.x;
  const int b   = blockIdx.y;

  const size_t tok0 = (size_t)b * SEQ + (size_t)seg * SEGT;
  const float* dsrc = ws_dt + tok0 * 32;
  float*       xdst = ws_xc + tok0 * 32;
  const float* zsrc = ws_z  + tok0 * 32;
  const float* bsrc = ws_B  + tok0 * 16;
  const float* csrc = ws_C  + tok0 * 16;

  ld_async16(&cdt[tid * 4], dsrc + tid * 4);
  ld_async16(&cxc[tid * 4], xdst + tid * 4);
  ld_async16(&cz[tid * 4],  zsrc + tid * 4);
  if (tid < 256) {
    ld_async16(&cB[tid * 4], bsrc + tid * 4);
    ld_async16(&cC[tid * 4], csrc + tid * 4);
  }
  const float A  = -__expf(A_log[d * 16 + s]);
  const float Dv = D_skip[d];
  float h = hinit[((size_t)b * SEGS + seg) * 512 + tid];
  wait_async_loads();
  __syncthreads();

#pragma unroll 4
  for (int t = 0; t < SEGT; ++t) {
    float dt = cdt[t * 32 + d];
    float xc = cxc[t * 32 + d];
    float dA = __expf(dt * A);
    h = dA * h + (dt * xc) * cB[t * 16 + s];
    float yp = h * cC[t * 16 + s];
    yp += __shfl_xor(yp, 1, 16);
    yp += __shfl_xor(yp, 2, 16);
    yp += __shfl_xor(yp, 4, 16);
    yp += __shfl_xor(yp, 8, 16);
    if (s == 0) {
      float yv = yp + Dv * xc;
      float zv = cz[t * 32 + d];
      cy[t * 32 + d] = yv * (zv / (1.f + __expf(-zv)));
    }
  }
  __syncthreads();
  st_async16(xdst + tid * 4, &cy[tid * 4]);   // ASYNCcnt store; ENDPGM waits idle
}

// ---------------------------------------------------------------------------
// Kernel 3: per 16-token tile (one wave):
//   out_proj GEMM + residual -> rmsnorm2 -> fc1/fc2/fc3 (relu) -> residual.
// ---------------------------------------------------------------------------
__global__ void __launch_bounds__(32)
k3_back(const float* __restrict__ ws_y, const float* __restrict__ x,
        const float* __restrict__ out_proj_w,  // [32][16]
        const float* __restrict__ rms2_w,
        const float* __restrict__ fc1_w, const float* __restrict__ fc1_b,
        const float* __restrict__ fc2_w, const float* __restrict__ fc2_b,
        const float* __restrict__ fc3_w, const float* __restrict__ fc3_b,
        float* __restrict__ out)
{
  __shared__ __align__(16) float ly[16][DI];
  __shared__ float lxn[16][16];
  __shared__ float lx2[16][16];
  __shared__ float lh1[16][HM];
  __shared__ float lh2[16][HM];

  const int lane = threadIdx.x;
  const int nn = lane & 15;
  const int hh = lane >> 4;
  const size_t t0 = (size_t)blockIdx.x * 16;

  const float* yt = ws_y + t0 * DI;
#pragma unroll
  for (int j = 0; j < 4; ++j)
    ld_async16(&ly[0][0] + j * 128 + lane * 4, yt + j * 128 + lane * 4);
  wait_async_loads();
  __syncthreads();

  // x1 = y @ out_proj_w ; xnew = x1 + x   (8 WMMAs)
  {
    v2f af[8];
#pragma unroll
    for (int kt = 0; kt < 8; ++kt) {
      int kk = kt * 4 + hh * 2;
      af[kt].x = ly[nn][kk]; af[kt].y = ly[nn][kk + 1];
    }
    v8f c = {0.f, 0.f, 0.f, 0.f, 0.f, 0.f, 0.f, 0.f};
#pragma unroll
    for (int kt = 0; kt < 8; ++kt) {
      int kr = kt * 4 + hh * 2;
      v2f bf;
      bf.x = out_proj_w[(kr + 0) * 16 + nn];
      bf.y = out_proj_w[(kr + 1) * 16 + nn];
      c = wmma4(af[kt], bf, c);
    }
#pragma unroll
    for (int r = 0; r < 8; ++r) {
      int M = r + 8 * hh;
      lxn[M][nn] = c[r] + x[(t0 + M) * DM + nn];
    }
  }
  __syncthreads();

  // rmsnorm2
  if (lane < 16) {
    float ss = 0.f;
#pragma unroll
    for (int cc = 0; cc < 16; ++cc) { float v = lxn[lane][cc]; ss += v * v; }
    float rstd = rsqrtf(ss * (1.0f / 16.0f) + EPS);
#pragma unroll
    for (int cc = 0; cc < 16; ++cc) lx2[lane][cc] = lxn[lane][cc] * rstd * rms2_w[cc];
  }
  __syncthreads();

  // h1 = relu(x2 @ fc1 + b1)   K=16, N=32  (8 WMMAs)
  {
    v2f af[4];
#pragma unroll
    for (int kt = 0; kt < 4; ++kt) {
      int kk = kt * 4 + hh * 2;
      af[kt].x = lx2[nn][kk]; af[kt].y = lx2[nn][kk + 1];
    }
#pragma unroll
    for (int nt = 0; nt < 2; ++nt) {
      v8f c = {0.f, 0.f, 0.f, 0.f, 0.f, 0.f, 0.f, 0.f};
#pragma unroll
      for (int kt = 0; kt < 4; ++kt) {
        int kr = kt * 4 + hh * 2;
        v2f bf;
        bf.x = fc1_w[(kr + 0) * HM + nt * 16 + nn];
        bf.y = fc1_w[(kr + 1) * HM + nt * 16 + nn];
        c = wmma4(af[kt], bf, c);
      }
      float bb = fc1_b[nt * 16 + nn];
#pragma unroll
      for (int r = 0; r < 8; ++r) lh1[r + 8 * hh][nt * 16 + nn] = fmaxf(c[r] + bb, 0.f);
    }
  }
  __syncthreads();

  // h2 = relu(h1 @ fc2 + b2)   K=32, N=32  (16 WMMAs)
  {
    v2f af[8];
#pragma unroll
    for (int kt = 0; kt < 8; ++kt) {
      int kk = kt * 4 + hh * 2;
      af[kt].x = lh1[nn][kk]; af[kt].y = lh1[nn][kk + 1];
    }
#pragma unroll
    for (int nt = 0; nt < 2; ++nt) {
      v8f c = {0.f, 0.f, 0.f, 0.f, 0.f, 0.f, 0.f, 0.f};
#pragma unroll
      for (int kt = 0; kt < 8; ++kt) {
        int kr = kt * 4 + hh * 2;
        v2f bf;
        bf.x = fc2_w[(kr + 0) * HM + nt * 16 + nn];
        bf.y = fc2_w[(kr + 1) * HM + nt * 16 + nn];
        c = wmma4(af[kt], bf, c);
      }
      float bb = fc2_b[nt * 16 + nn];
#pragma unroll
      for (int r = 0; r < 8; ++r) lh2[r + 8 * hh][nt * 16 + nn] = fmaxf(c[r] + bb, 0.f);
    }
  }
  __syncthreads();

  // x3 = h2 @ fc3 + b3 ; out = xnew + x3   (8 WMMAs)
  {
    v2f af[8];
#pragma unroll
    for (int kt = 0; kt < 8; ++kt) {
      int kk = kt * 4 + hh * 2;
      af[kt].x = lh2[nn][kk]; af[kt].y = lh2[nn][kk + 1];
    }
    v8f c = {0.f, 0.f, 0.f, 0.f, 0.f, 0.f, 0.f, 0.f};
#pragma unroll
    for (int kt = 0; kt < 8; ++kt) {
      int kr = kt * 4 + hh * 2;
      v2f bf;
      bf.x = fc3_w[(kr + 0) * 16 + nn];
      bf.y = fc3_w[(kr + 1) * 16 + nn];
      c = wmma4(af[kt], bf, c);
    }
    float bb = fc3_b[nn];
#pragma unroll
    for (int r = 0; r < 8; ++r) {
      int M = r + 8 * hh;
      out[(t0 + M) * DM + nn] = lxn[M][nn] + c[r] + bb;
    }
  }
}

// ---------------------------------------------------------------------------
extern "C" void kernel_launch(void* const* d_in, const int* in_sizes, int n_in,
                              void* d_out, int out_size, void* d_ws, size_t ws_size,
                              hipStream_t stream) {
  (void)in_sizes; (void)n_in; (void)out_size; (void)ws_size;
  const float* x        = (const float*)d_in[0];
  const float* rms1_w   = (const float*)d_in[1];
  const float* in_proj  = (const float*)d_in[2];
  const float* conv_w   = (const float*)d_in[3];
  const float* conv_b   = (const float*)d_in[4];
  const float* x_proj   = (const float*)d_in[5];
  const float* dt_w     = (const float*)d_in[6];
  const float* dt_b     = (const float*)d_in[7];
  const float* A_log    = (const float*)d_in[8];
  const float* D_skip   = (const float*)d_in[9];
  const float* out_proj = (const float*)d_in[10];
  const float* rms2_w   = (const float*)d_in[11];
  const float* fc1_w    = (const float*)d_in[12];
  const float* fc1_b    = (const float*)d_in[13];
  const float* fc2_w    = (const float*)d_in[14];
  const float* fc2_b    = (const float*)d_in[15];
  const float* fc3_w    = (const float*)d_in[16];
  const float* fc3_b    = (const float*)d_in[17];
  float* out = (float*)d_out;

  const size_t NT = (size_t)BATCH * SEQ;    // 131072 tokens
  const size_t NCH = (size_t)BATCH * SEGS * 512;  // per-segment chain records
  float* ws    = (float*)d_ws;
  float* wz    = ws;                   // [NT][32]
  float* wxc   = ws + NT * 32;         // [NT][32]  (scan overwrites with gated y)
  float* wdt   = ws + NT * 64;         // [NT][32]
  float* wB    = ws + NT * 96;         // [NT][16]
  float* wC    = ws + NT * 112;        // [NT][16]
  float* wsegA = ws + NT * 128;        // [B][SEGS][512]
  float* wsegB = wsegA + NCH;          // [B][SEGS][512]
  float* whini = wsegB + NCH;          // [B][SEGS][512]

  const int TILES = (int)(NT / 16);    // 8192 tiles of 16 tokens

  hipLaunchKernelGGL(k1_front, dim3(TILES), dim3(32), 0, stream,
                     x, rms1_w, in_proj, conv_w, conv_b, x_proj, dt_w, dt_b,
                     wz, wxc, wdt, wB, wC);
  hipLaunchKernelGGL(k2a_segscan, dim3(SEGS, BATCH), dim3(512), 0, stream,
                     wB, wdt, wxc, A_log, wsegA, wsegB);
  hipLaunchKernelGGL(k2b_stitch, dim3(BATCH), dim3(512), 0, stream,
                     wsegA, wsegB, whini);
  hipLaunchKernelGGL(k2c_apply, dim3(SEGS, BATCH), dim3(512), 0, stream,
                     wB, wC, wdt, wxc, wz, A_log, D_skip, whini);
  hipLaunchKernelGGL(k3_back, dim3(TILES), dim3(32), 0, stream,
                     wxc, x, out_proj, rms2_w, fc1_w, fc1_b, fc2_w, fc2_b,
                     fc3_w, fc3_b, out);
}